// OutConv_83657372991769
// MI455X (gfx1250) — compile-verified
//
#include <hip/hip_runtime.h>
#include <hip/hip_bf16.h>
#include <math.h>

typedef __attribute__((ext_vector_type(2))) float v2f;
typedef __attribute__((ext_vector_type(8))) float v8f;

#define PI2 6.28318530717958647692f

// ---------------------------------------------------------------------------
// 256-point Stockham radix-2 FFT over LDS ping-pong buffers.
// Each warp processes `nrows` independent 256-point sequences.
// Sequence rr: element i lives at buf[(row0+rr)*rstride + i*estride].
// sign = -1 forward, +1 inverse (unscaled). 8 stages -> result back in bufA.
// ---------------------------------------------------------------------------
__device__ __forceinline__ void fft256_batch(float2* bufA, float2* bufB,
                                             int lane, int row0, int nrows,
                                             int rstride, int estride,
                                             float sign)
{
  float2* src = bufA;
  float2* dst = bufB;
  int n = 256, s = 1;
  #pragma unroll
  for (int stage = 0; stage < 8; ++stage) {
    const int   m     = n >> 1;
    const float theta = sign * PI2 / (float)n;
    for (int rr = 0; rr < nrows; ++rr) {
      const int base = (row0 + rr) * rstride;
      #pragma unroll
      for (int j = 0; j < 4; ++j) {
        const int t = lane + 32 * j;        // 128 butterflies / 32 lanes
        const int p = t / s;
        const int q = t - p * s;
        float si, co;
        __sincosf(theta * (float)p, &si, &co);
        const float2 a  = src[base + (q + s * p) * estride];
        const float2 b  = src[base + (q + s * (p + m)) * estride];
        const float2 su = make_float2(a.x + b.x, a.y + b.y);
        const float2 di = make_float2(a.x - b.x, a.y - b.y);
        const float2 tw = make_float2(di.x * co - di.y * si,
                                      di.x * si + di.y * co);
        dst[base + (q + 2 * s * p) * estride]     = su;
        dst[base + (q + 2 * s * p + s) * estride] = tw;
      }
    }
    float2* tmp = src; src = dst; dst = tmp;
    n >>= 1;
    s <<= 1;
    __syncthreads();
  }
}

// ---------------------------------------------------------------------------
// Kernel 1: fused 1x1-conv (WMMA f32 GEMM) + forward row FFT.
// One block per (b, h): GEMM  y[w][o] = sum_c x[c][w] * W[o][c] + bias[o]
// with M=256 (w), N=32 (o), K=64 (c) using V_WMMA_F32_16X16X4_F32,
// then 32 row-FFTs of length 256, complex output to yfft[(b*32+o)][h][w].
// ---------------------------------------------------------------------------
__global__ __launch_bounds__(256)
void conv_rowfft_kernel(const float* __restrict__ xg,
                        const float* __restrict__ wg,
                        const float* __restrict__ bg,
                        float2* __restrict__ yfft)
{
  __shared__ float s_x[64 * 256];   // xs[c][w]; reused as FFT ping-pong bufs
  __shared__ float s_wt[64 * 33];   // wt[c][o], padded stride 33
  __shared__ float s_b[32];

  const int tid  = threadIdx.x;
  const int lane = tid & 31;
  const int wv   = tid >> 5;        // 8 waves
  const int bid  = blockIdx.x;
  const int b    = bid >> 8;
  const int h    = bid & 255;

  // weights transposed into LDS: wt[c][o]
  for (int i = tid; i < 32 * 64; i += 256) {
    const int o = i >> 6, c = i & 63;
    s_wt[c * 33 + o] = wg[i];
  }
  if (tid < 32) s_b[tid] = bg[tid];

  // x[b, :, h, :] -> LDS (64 rows of 256 f32, coalesced)
  const float* xrow = xg + (((size_t)b * 64) * 256 + h) * 256;
  for (int c = 0; c < 64; ++c) {
    __builtin_prefetch(xrow + (size_t)(c + 1) * 65536 + tid, 0, 0);
    s_x[c * 256 + tid] = xrow[(size_t)c * 65536 + tid];
  }
  __syncthreads();

  // ---- GEMM via V_WMMA_F32_16X16X4_F32 --------------------------------
  // Wave wv owns M-tiles {2wv, 2wv+1} x N-tiles {0,1}; K loop of 16 steps.
  v8f acc[2][2] = {};
  const int half = lane >> 4;   // 0: lanes 0-15, 1: lanes 16-31
  const int l15  = lane & 15;
  #pragma unroll
  for (int kk = 0; kk < 16; ++kk) {
    const int kb = kk * 4 + 2 * half;           // K base for this lane half
    v2f afrag[2], bfrag[2];
    #pragma unroll
    for (int i = 0; i < 2; ++i) {               // A: 16x4 f32 layout
      const int m = (wv * 2 + i) * 16 + l15;    // row M = w position
      afrag[i][0] = s_x[kb * 256 + m];          // VGPR0: K=kb
      afrag[i][1] = s_x[(kb + 1) * 256 + m];    // VGPR1: K=kb+1
    }
    #pragma unroll
    for (int jb = 0; jb < 2; ++jb) {            // B: 4x16 f32 layout
      const int n = jb * 16 + l15;              // col N = out channel
      bfrag[jb][0] = s_wt[kb * 33 + n];
      bfrag[jb][1] = s_wt[(kb + 1) * 33 + n];
    }
    #pragma unroll
    for (int i = 0; i < 2; ++i)
      #pragma unroll
      for (int jb = 0; jb < 2; ++jb)
        acc[i][jb] = __builtin_amdgcn_wmma_f32_16x16x4_f32(
            false, afrag[i], false, bfrag[jb], (short)0, acc[i][jb],
            false, false);
  }
  __syncthreads();              // done with s_x as GEMM operand

  // ---- bias + forward row FFT, 16 channels at a time ------------------
  float2* bufA = reinterpret_cast<float2*>(s_x);   // 16 rows x 256 complex
  float2* bufB = bufA + 16 * 256;

  for (int hf = 0; hf < 2; ++hf) {
    // scatter C tiles (16x16 f32 layout: VGPR r -> M=r+8*half, N=l15)
    #pragma unroll
    for (int i = 0; i < 2; ++i) {
      #pragma unroll
      for (int r = 0; r < 8; ++r) {
        const int   w  = (wv * 2 + i) * 16 + r + 8 * half;
        const int   oc = l15;                     // channel within half
        const float v  = acc[i][hf][r] + s_b[hf * 16 + oc];
        bufA[oc * 256 + w] = make_float2(v, 0.0f);
      }
    }
    __syncthreads();
    // 16 row-FFTs; each warp handles rows 2wv, 2wv+1
    fft256_batch(bufA, bufB, lane, wv * 2, 2, 256, 1, -1.0f);
    // write complex rows to global: yfft[(b*32 + hf*16 + oc)][h][:]
    for (int oc = 0; oc < 16; ++oc) {
      yfft[(((size_t)b * 32 + hf * 16 + oc) * 256 + h) * 256 + tid] =
          bufA[oc * 256 + tid];
    }
    __syncthreads();
  }
}

// ---------------------------------------------------------------------------
// Kernel 2: column FFT + highpass mask + column IFFT, fused per 16-col tile.
// ---------------------------------------------------------------------------
__global__ __launch_bounds__(256)
void colfft_mask_kernel(float2* __restrict__ yfft)
{
  __shared__ float2 bufA[256 * 16];   // [h][col], col fastest
  __shared__ float2 bufB[256 * 16];

  const int tid  = threadIdx.x;
  const int lane = tid & 31;
  const int wv   = tid >> 5;
  const int img  = blockIdx.y;          // 0..511  (b*32+o)
  const int w0   = blockIdx.x * 16;     // column tile base
  float2* base = yfft + (size_t)img * 65536;

  for (int i = 0; i < 16; ++i) {
    const int idx = i * 256 + tid;      // idx = hh*16 + col
    const int hh = idx >> 4, col = idx & 15;
    bufA[idx] = base[hh * 256 + w0 + col];
  }
  __syncthreads();

  // forward FFT along h (stride-16 elements); each warp: cols 2wv, 2wv+1
  fft256_batch(bufA, bufB, lane, wv * 2, 2, 1, 16, -1.0f);

  // highpass mask: k_i = fftfreq(256)[i] * 256*2pi/25 = sgn(i)*|i|*2pi/25
  const float kfac = PI2 / 25.0f;
  for (int i = 0; i < 16; ++i) {
    const int idx = i * 256 + tid;
    const int kh  = idx >> 4;
    const int kw  = w0 + (idx & 15);
    const float fy = (float)((kh < 128) ? kh : kh - 256) * kfac;
    const float fx = (float)((kw < 128) ? kw : kw - 256) * kfac;
    const float k  = sqrtf(fx * fx + fy * fy);
    float msk = __expf(-3.0f / (k + 1e-8f));
    if (kh == 0 && kw == 0) msk = 0.0f;
    const float2 v = bufA[idx];
    bufA[idx] = make_float2(v.x * msk, v.y * msk);
  }
  __syncthreads();

  // inverse FFT along h (unscaled; 1/N^2 applied in final kernel)
  fft256_batch(bufA, bufB, lane, wv * 2, 2, 1, 16, 1.0f);

  for (int i = 0; i < 16; ++i) {
    const int idx = i * 256 + tid;
    const int hh = idx >> 4, col = idx & 15;
    base[hh * 256 + w0 + col] = bufA[idx];
  }
}

// ---------------------------------------------------------------------------
// Kernel 3: inverse row FFT, take real part, apply 1/(256*256) norm.
// ---------------------------------------------------------------------------
__global__ __launch_bounds__(256)
void rowifft_kernel(const float2* __restrict__ yfft, float* __restrict__ out)
{
  __shared__ float2 bufA[8 * 256];
  __shared__ float2 bufB[8 * 256];

  const int tid  = threadIdx.x;
  const int lane = tid & 31;
  const int wv   = tid >> 5;
  const size_t rowBase = (size_t)blockIdx.x * 8;   // 8 rows per block

  for (int i = 0; i < 8; ++i)
    bufA[i * 256 + tid] = yfft[(rowBase + i) * 256 + tid];
  __syncthreads();

  fft256_batch(bufA, bufB, lane, wv, 1, 256, 1, 1.0f);  // one row per warp

  const float sc = 1.0f / 65536.0f;
  for (int i = 0; i < 8; ++i)
    out[(rowBase + i) * 256 + tid] = bufA[i * 256 + tid].x * sc;
}

// ---------------------------------------------------------------------------
extern "C" void kernel_launch(void* const* d_in, const int* in_sizes, int n_in,
                              void* d_out, int out_size, void* d_ws,
                              size_t ws_size, hipStream_t stream)
{
  const float* x = (const float*)d_in[0];   // [16,64,256,256]
  const float* W = (const float*)d_in[1];   // [32,64]
  const float* b = (const float*)d_in[2];   // [32]
  float*  out  = (float*)d_out;             // [16,32,256,256]
  float2* yfft = (float2*)d_ws;             // 512*256*256 complex = 256 MiB

  conv_rowfft_kernel<<<dim3(16 * 256), dim3(256), 0, stream>>>(x, W, b, yfft);
  colfft_mask_kernel<<<dim3(16, 512), dim3(256), 0, stream>>>(yfft);
  rowifft_kernel<<<dim3(512 * 256 / 8), dim3(256), 0, stream>>>(yfft, out);
}